// TransformerDecoderWithCache_13709535609488
// MI455X (gfx1250) — compile-verified
//
#include <hip/hip_runtime.h>
#include <hip/hip_bf16.h>
#include <cstddef>

// ---------------------------------------------------------------------------
// Transformer decoder step with KV cache, MI455X (gfx1250, wave32).
// Bandwidth-bound (~1.4GB traffic, ~2.3GFLOP): weights (604MB) streamed
// through v_wmma_f32_16x16x4_f32 mat-vec tiles with pointer-bumped
// addressing + WGP-scope prefetch; cache shift (403MB r + 403MB w) staged
// through LDS via GLOBAL_LOAD_ASYNC_TO_LDS_B128 (ASYNCcnt) when available.
// Sequential layer dependency expressed as stream-ordered launches.
// ---------------------------------------------------------------------------

#define L_  24
#define E_  1024
#define H_  16
#define HD_ 64
#define S_  2048

#define NEG_INF (-__builtin_huge_valf())

typedef __attribute__((ext_vector_type(2))) float v2f;
typedef __attribute__((ext_vector_type(8))) float v8f;

#if defined(__AMDGCN__) && __has_builtin(__builtin_amdgcn_global_load_async_to_lds_b128)
#define HAVE_ASYNC_LDS 1
// Builtin param types (from clang diagnostic): pointer to GCC-vector int4,
// AS1 (global) for the source, AS3 (LDS) for the destination.
typedef int gi4 __attribute__((vector_size(16)));
typedef __attribute__((address_space(1))) gi4 AS1_i4;
typedef __attribute__((address_space(3))) gi4 AS3_i4;
#else
#define HAVE_ASYNC_LDS 0
#endif

__device__ __forceinline__ void wait_async_zero() {
#if HAVE_ASYNC_LDS
#if __has_builtin(__builtin_amdgcn_s_wait_asynccnt)
  __builtin_amdgcn_s_wait_asynccnt(0);
#else
  asm volatile("s_wait_asynccnt 0x0" ::: "memory");
#endif
#endif
}

__device__ __forceinline__ float wave_sum(float v) {
  #pragma unroll
  for (int m = 16; m > 0; m >>= 1) v += __shfl_xor(v, m, 32);
  return v;
}
__device__ __forceinline__ float wave_max(float v) {
  #pragma unroll
  for (int m = 16; m > 0; m >>= 1) v = fmaxf(v, __shfl_xor(v, m, 32));
  return v;
}

// ---------------------------------------------------------------------------
// Shift-copy: dst[l][s] = src[l][s+1] for s in [0, S-1).  One block per row,
// 256 threads x float4 = 1024 floats.  Data staged global->LDS with the
// async-DMA path (ASYNCcnt) when the toolchain exposes it.  Optionally emits
// mask[l*S+s] = any(row != 0) (attention masking of the v-cache).
// ---------------------------------------------------------------------------
__global__ void shift_copy_kernel(const float* __restrict__ src,
                                  float* __restrict__ dst,
                                  int* __restrict__ mask) {
  int r = blockIdx.x;                    // 0 .. L*(S-1)-1
  int l = r / (S_ - 1);
  int s = r % (S_ - 1);
  const float4* sp = (const float4*)(src + ((size_t)l * S_ + s + 1) * E_);
  float4* dp = (float4*)(dst + ((size_t)l * S_ + s) * E_);
  float4 d;
#if HAVE_ASYNC_LDS
  __shared__ float4 stage[256];
  // async copy of 16B/lane into this block's LDS tile (ASYNCcnt-tracked)
  __builtin_amdgcn_global_load_async_to_lds_b128(
      (AS1_i4*)(sp + threadIdx.x), (AS3_i4*)&stage[threadIdx.x], 0, 0);
  wait_async_zero();
  __syncthreads();
  d = stage[threadIdx.x];
#else
  d = sp[threadIdx.x];
#endif
  dp[threadIdx.x] = d;
  if (mask != nullptr) {
    __shared__ int sh;
    if (threadIdx.x == 0) sh = 0;
    __syncthreads();
    int nz = (d.x != 0.f) | (d.y != 0.f) | (d.z != 0.f) | (d.w != 0.f);
    if (nz) atomicOr(&sh, 1);
    __syncthreads();
    if (threadIdx.x == 0) mask[(size_t)l * S_ + s] = sh;
  }
}

// ---------------------------------------------------------------------------
// Embedding gather + layernorm(eps=1e-12).  1 block x 256 threads, E=1024.
// ---------------------------------------------------------------------------
__global__ void embed_ln_kernel(const int* __restrict__ x,
                                const float* __restrict__ et,
                                const float* __restrict__ ei,
                                const float* __restrict__ ex,
                                const float* __restrict__ ey,
                                const float* __restrict__ etm,
                                const float* __restrict__ sc,
                                const float* __restrict__ bi,
                                float* __restrict__ h) {
  __shared__ float rs[8], rq[8], bm[2];
  int t = threadIdx.x, lane = t & 31, wave = t >> 5;
  const float* p0 = et  + (size_t)x[0] * E_;
  const float* p1 = ei  + (size_t)x[1] * E_;
  const float* p2 = ex  + (size_t)x[2] * E_;
  const float* p3 = ey  + (size_t)x[3] * E_;
  const float* p4 = etm + (size_t)x[4] * E_;
  float v[4], sum = 0.f, sq = 0.f;
  #pragma unroll
  for (int j = 0; j < 4; ++j) {
    int idx = t * 4 + j;
    float y = p0[idx] + p1[idx] + p2[idx] + p3[idx] + p4[idx];
    v[j] = y; sum += y; sq += y * y;
  }
  sum = wave_sum(sum); sq = wave_sum(sq);
  if (lane == 0) { rs[wave] = sum; rq[wave] = sq; }
  __syncthreads();
  if (wave == 0) {
    float a = (lane < 8) ? rs[lane] : 0.f;
    float c = (lane < 8) ? rq[lane] : 0.f;
    a = wave_sum(a); c = wave_sum(c);
    if (lane == 0) {
      float mean = a * (1.0f / E_);
      float var  = c * (1.0f / E_) - mean * mean;
      bm[0] = mean; bm[1] = rsqrtf(var + 1e-12f);
    }
  }
  __syncthreads();
  float mean = bm[0], inv = bm[1];
  #pragma unroll
  for (int j = 0; j < 4; ++j) {
    int idx = t * 4 + j;
    h[idx] = (v[j] - mean) * inv * sc[idx] + bi[idx];
  }
}

// ---------------------------------------------------------------------------
// Mat-vec via WMMA:  y[n] = act( sum_e x[e]*W[e*N+n] + b[n] ).
// x staged once per block into LDS (frees global-load slots for W stream).
// One wave owns a 16-column tile.  Per K-step of 4:
//   A(16x4) = broadcast of xs[e0..e0+3] across rows, B(4x16) = W tile.
// A layout: lane<16 holds K={0,1}, lane>=16 holds K={2,3} (M = lane&15).
// B layout: VGPR0 = K0/K2 rows, VGPR1 = K1/K3 rows, N = lane&15.
// D row 0 (c[0], lanes 0..15) = the 16 outputs.
// Pointer-bumped addressing (no in-loop 64-bit muls), unconditional
// speculative prefetch 2 steps ahead at WGP scope.
// ---------------------------------------------------------------------------
__global__ void matvec_wmma_kernel(const float* __restrict__ x,
                                   const float* __restrict__ W,
                                   const float* __restrict__ b,
                                   float* __restrict__ y,
                                   int K, int N, int relu) {
  __shared__ float xs[E_];
  for (int j = threadIdx.x; j < K; j += blockDim.x) xs[j] = x[j];
  __syncthreads();

  int lane = threadIdx.x & 31;
  int wave = threadIdx.x >> 5;
  int tile = blockIdx.x * (blockDim.x >> 5) + wave;
  if (tile * 16 >= N) return;               // uniform per wave; after barrier
  int n0 = tile * 16;
  int kb = (lane < 16) ? 0 : 2;
  int nc = n0 + (lane & 15);

  const size_t rowN = (size_t)N;
  const size_t step = rowN * 4;             // 4 K-rows per WMMA
  const float* wp = W + (size_t)kb * rowN + nc;
  v8f c = {};
  int iters = K >> 2;
  #pragma unroll 4
  for (int it = 0; it < iters; ++it) {
    v2f a, bt;
    a.x  = xs[(it << 2) + kb];
    a.y  = xs[(it << 2) + kb + 1];
    bt.x = wp[0];
    bt.y = wp[rowN];
    __builtin_prefetch(wp + 2 * step, 0, 3);   // speculative, WGP scope
    c = __builtin_amdgcn_wmma_f32_16x16x4_f32(
        /*neg_a=*/false, a, /*neg_b=*/false, bt,
        /*c_mod=*/(short)0, c, /*reuse_a=*/false, /*reuse_b=*/false);
    wp += step;
  }
  if (lane < 16) {
    float r = c[0] + b[n0 + lane];
    if (relu) r = fmaxf(r, 0.f);
    y[n0 + lane] = r;
  }
}

// ---------------------------------------------------------------------------
// mask for the newly appended v row: mask[S-1] = any(v_new != 0)
// ---------------------------------------------------------------------------
__global__ void mask_last_kernel(const float* __restrict__ vrow,
                                 int* __restrict__ flag) {
  __shared__ int sh;
  if (threadIdx.x == 0) sh = 0;
  __syncthreads();
  int nz = 0;
  for (int j = threadIdx.x; j < E_; j += 256) nz |= (vrow[j] != 0.0f);
  if (nz) atomicOr(&sh, 1);
  __syncthreads();
  if (threadIdx.x == 0) flag[0] = sh;
}

// ---------------------------------------------------------------------------
// Attention scores: att[h][s] = mask[s] ? 0.125 * dot64(q[h], k[s][h]) : -inf
// One wave per (h,s); lanes cover d and d+32, shuffle-reduce.
// ---------------------------------------------------------------------------
__global__ void attn_score_kernel(const float* __restrict__ q,
                                  const float* __restrict__ k,   // [S][H][HD]
                                  const int* __restrict__ mask,  // [S]
                                  float* __restrict__ att) {     // [H][S]
  int w = blockIdx.x * (blockDim.x >> 5) + (threadIdx.x >> 5);
  int lane = threadIdx.x & 31;
  int head = w >> 11;            // w / S_  (S_ = 2048)
  int s = w & (S_ - 1);
  const float* kp = k + (size_t)s * E_ + head * HD_;
  const float* qp = q + head * HD_;
  float p = qp[lane] * kp[lane] + qp[lane + 32] * kp[lane + 32];
  p = wave_sum(p);
  if (lane == 0) {
    att[(size_t)head * S_ + s] = mask[s] ? p * 0.125f : NEG_INF;
  }
}

// ---------------------------------------------------------------------------
// Softmax over S per head.  One block (256 thr) per head.
// ---------------------------------------------------------------------------
__global__ void softmax_rows_kernel(float* __restrict__ att) {
  __shared__ float red[8];
  __shared__ float bcast;
  float* row = att + (size_t)blockIdx.x * S_;
  int lane = threadIdx.x & 31, wave = threadIdx.x >> 5;

  float m = NEG_INF;
  for (int s = threadIdx.x; s < S_; s += 256) m = fmaxf(m, row[s]);
  m = wave_max(m);
  if (lane == 0) red[wave] = m;
  __syncthreads();
  if (wave == 0) {
    float t = red[lane & 7];
    t = wave_max(t);
    if (lane == 0) bcast = t;
  }
  __syncthreads();
  m = bcast;

  float sum = 0.f;
  for (int s = threadIdx.x; s < S_; s += 256) sum += __expf(row[s] - m);
  sum = wave_sum(sum);
  if (lane == 0) red[wave] = sum;
  __syncthreads();
  if (wave == 0) {
    float t = (lane < 8) ? red[lane] : 0.f;
    t = wave_sum(t);
    if (lane == 0) bcast = t;
  }
  __syncthreads();
  float inv = 1.0f / bcast;
  for (int s = threadIdx.x; s < S_; s += 256) row[s] = __expf(row[s] - m) * inv;
}

// ---------------------------------------------------------------------------
// vals[h][d] = sum_s att[h][s] * v[s][h][d].  Block per head, 256 threads =
// 4 s-partitions x 64 d-lanes, shared-mem tree combine.
// ---------------------------------------------------------------------------
__global__ void attn_av_kernel(const float* __restrict__ att,
                               const float* __restrict__ v,   // [S][H][HD]
                               float* __restrict__ vals) {    // [H*HD]
  __shared__ float acc_sh[256];
  int head = blockIdx.x;
  int d  = threadIdx.x & 63;
  int sp = threadIdx.x >> 6;   // 0..3
  const float* row = att + (size_t)head * S_;
  float acc = 0.f;
  for (int s = sp; s < S_; s += 4)
    acc = fmaf(row[s], v[(size_t)s * E_ + head * HD_ + d], acc);
  acc_sh[threadIdx.x] = acc;
  __syncthreads();
  if (sp == 0) {
    float r = acc_sh[d] + acc_sh[64 + d] + acc_sh[128 + d] + acc_sh[192 + d];
    vals[head * HD_ + d] = r;
  }
}

// ---------------------------------------------------------------------------
// h = layernorm(h + r, sc, bi, eps).  1 block x 256 threads.
// ---------------------------------------------------------------------------
__global__ void add_ln_kernel(float* __restrict__ h,
                              const float* __restrict__ r,
                              const float* __restrict__ sc,
                              const float* __restrict__ bi,
                              float eps) {
  __shared__ float rs[8], rq[8], bm[2];
  int t = threadIdx.x, lane = t & 31, wave = t >> 5;
  float v[4], sum = 0.f, sq = 0.f;
  #pragma unroll
  for (int j = 0; j < 4; ++j) {
    int idx = t * 4 + j;
    float y = h[idx] + r[idx];
    v[j] = y; sum += y; sq += y * y;
  }
  sum = wave_sum(sum); sq = wave_sum(sq);
  if (lane == 0) { rs[wave] = sum; rq[wave] = sq; }
  __syncthreads();
  if (wave == 0) {
    float a = (lane < 8) ? rs[lane] : 0.f;
    float c = (lane < 8) ? rq[lane] : 0.f;
    a = wave_sum(a); c = wave_sum(c);
    if (lane == 0) {
      float mean = a * (1.0f / E_);
      float var  = c * (1.0f / E_) - mean * mean;
      bm[0] = mean; bm[1] = rsqrtf(var + eps);
    }
  }
  __syncthreads();
  float mean = bm[0], inv = bm[1];
  #pragma unroll
  for (int j = 0; j < 4; ++j) {
    int idx = t * 4 + j;
    h[idx] = (v[j] - mean) * inv * sc[idx] + bi[idx];
  }
}

// ---------------------------------------------------------------------------
// Output heads: 47 small dots (32 + 7 + 8).  1 block, wave per output.
// ---------------------------------------------------------------------------
__global__ void heads_kernel(const float* __restrict__ h,
                             const float* __restrict__ Wpi, const float* __restrict__ bpi,
                             const float* __restrict__ Wvh, const float* __restrict__ bvh,
                             const float* __restrict__ Wc,  const float* __restrict__ bc,
                             float* __restrict__ out) {
  int wave = threadIdx.x >> 5, lane = threadIdx.x & 31;
  for (int o = wave; o < 47; o += 8) {
    const float* W; const float* b; int col, ncol;
    if (o < 32)      { W = Wpi; b = bpi; col = o;      ncol = 32; }
    else if (o < 39) { W = Wvh; b = bvh; col = o - 32; ncol = 7;  }
    else             { W = Wc;  b = bc;  col = o - 39; ncol = 8;  }
    float acc = 0.f;
    for (int e = lane; e < E_; e += 32) acc += h[e] * W[(size_t)e * ncol + col];
    acc = wave_sum(acc);
    if (lane == 0) out[o] = acc + b[col];
  }
}

// ---------------------------------------------------------------------------
// Host-side orchestration (all on `stream`, graph-capture safe).
// ---------------------------------------------------------------------------
extern "C" void kernel_launch(void* const* d_in, const int* in_sizes, int n_in,
                              void* d_out, int out_size, void* d_ws, size_t ws_size,
                              hipStream_t stream) {
  const int*   x        = (const int*)  d_in[0];
  const float* cache1   = (const float*)d_in[1];
  const float* cache2   = (const float*)d_in[2];
  const float* emb_type = (const float*)d_in[3];
  const float* emb_id   = (const float*)d_in[4];
  const float* emb_x    = (const float*)d_in[5];
  const float* emb_y    = (const float*)d_in[6];
  const float* emb_t    = (const float*)d_in[7];
  const float* ln_emb_s = (const float*)d_in[8];
  const float* ln_emb_b = (const float*)d_in[9];
  const float* Wv = (const float*)d_in[10];
  const float* Wq = (const float*)d_in[11];
  const float* Wk = (const float*)d_in[12];
  const float* Wo = (const float*)d_in[13];
  const float* W1 = (const float*)d_in[14];
  const float* W2 = (const float*)d_in[15];
  const float* bv = (const float*)d_in[16];
  const float* bq = (const float*)d_in[17];
  const float* bk = (const float*)d_in[18];
  const float* bo = (const float*)d_in[19];
  const float* b1 = (const float*)d_in[20];
  const float* b2 = (const float*)d_in[21];
  const float* ln1_s = (const float*)d_in[22];
  const float* ln1_b = (const float*)d_in[23];
  const float* ln2_s = (const float*)d_in[24];
  const float* ln2_b = (const float*)d_in[25];
  const float* W_pi = (const float*)d_in[26];
  const float* b_pi = (const float*)d_in[27];
  const float* W_vh = (const float*)d_in[28];
  const float* b_vh = (const float*)d_in[29];
  const float* W_c  = (const float*)d_in[30];
  const float* b_c  = (const float*)d_in[31];

  float* out = (float*)d_out;
  const size_t cache_elems = (size_t)L_ * S_ * E_;   // 50,331,648
  float* nc1 = out + 47;                 // next_cache1
  float* nc2 = nc1 + cache_elems;        // next_cache2

  float* ws   = (float*)d_ws;
  float* h    = ws;                        // [E]
  float* q    = ws + 1024;                 // [E]
  float* vals = ws + 2048;                 // [E]
  float* t1   = ws + 3072;                 // [E]
  float* t2   = ws + 4096;                 // [E]
  float* att  = ws + 5120;                 // [H*S] = 32768
  int*   mask = (int*)(ws + 5120 + (size_t)H_ * S_);   // [L*S] ints

  const int rows = L_ * (S_ - 1);

  // Bulk cache shift (off the critical path, saturates HBM in parallel).
  shift_copy_kernel<<<rows, 256, 0, stream>>>(cache2, nc2, nullptr);
  shift_copy_kernel<<<rows, 256, 0, stream>>>(cache1, nc1, mask);

  // h0 = LN(sum of embeddings)
  embed_ln_kernel<<<1, 256, 0, stream>>>(x, emb_type, emb_id, emb_x, emb_y,
                                         emb_t, ln_emb_s, ln_emb_b, h);

  const size_t WSZ = (size_t)E_ * E_;
  for (int i = 0; i < L_; ++i) {
    float* vrow = nc1 + ((size_t)i * S_ + (S_ - 1)) * E_;
    float* krow = nc2 + ((size_t)i * S_ + (S_ - 1)) * E_;
    int*   mrow = mask + (size_t)i * S_;

    // v, q, k projections (WMMA mat-vecs); v/k land directly in next_cache.
    matvec_wmma_kernel<<<8, 256, 0, stream>>>(h, Wv + i * WSZ, bv + i * E_, vrow, E_, E_, 0);
    matvec_wmma_kernel<<<8, 256, 0, stream>>>(h, Wq + i * WSZ, bq + i * E_, q,    E_, E_, 0);
    matvec_wmma_kernel<<<8, 256, 0, stream>>>(h, Wk + i * WSZ, bk + i * E_, krow, E_, E_, 0);
    mask_last_kernel<<<1, 256, 0, stream>>>(vrow, mrow + (S_ - 1));

    // attention over the (already shifted) caches
    attn_score_kernel<<<(H_ * S_) / 8, 256, 0, stream>>>(
        q, nc2 + (size_t)i * S_ * E_, mrow, att);
    softmax_rows_kernel<<<H_, 256, 0, stream>>>(att);
    attn_av_kernel<<<H_, 256, 0, stream>>>(att, nc1 + (size_t)i * S_ * E_, vals);

    // output proj + LN1
    matvec_wmma_kernel<<<8, 256, 0, stream>>>(vals, Wo + i * WSZ, bo + i * E_, t1, E_, E_, 0);
    add_ln_kernel<<<1, 256, 0, stream>>>(h, t1, ln1_s + i * E_, ln1_b + i * E_, 1e-6f);

    // FFN + LN2
    matvec_wmma_kernel<<<8, 256, 0, stream>>>(h,  W1 + i * WSZ, b1 + i * E_, t1, E_, E_, 1);
    matvec_wmma_kernel<<<8, 256, 0, stream>>>(t1, W2 + i * WSZ, b2 + i * E_, t2, E_, E_, 0);
    add_ln_kernel<<<1, 256, 0, stream>>>(h, t2, ln2_s + i * E_, ln2_b + i * E_, 1e-6f);
  }

  heads_kernel<<<1, 256, 0, stream>>>(h, W_pi, b_pi, W_vh, b_vh, W_c, b_c, out);
}